// Context_UpsampleLayer_6047313953115
// MI455X (gfx1250) — compile-verified
//
#include <hip/hip_runtime.h>
#include <hip/hip_bf16.h>
#include <math.h>

typedef float v2f __attribute__((ext_vector_type(2)));
typedef float v8f __attribute__((ext_vector_type(8)));

// ---------------------------------------------------------------------------
// Helper kernels
// ---------------------------------------------------------------------------

// Fill the inverse neighbor map with the padding index n (row n of padded
// feature buffers is kept at zero).
__global__ void init_gidx_kernel(int* __restrict__ gidx, int total, int n) {
    int i = blockIdx.x * blockDim.x + threadIdx.x;
    if (i < total) gidx[i] = n;
}

// Zero the padding row of every padded feature buffer.
__global__ void zero_pads_kernel(float* fup, float* fa, float* fb,
                                 float* t0, float* t1, float* t2, int n) {
    int t = threadIdx.x;
    if (t < 32) {
        fup[(size_t)n * 32 + t] = 0.f;
        fa [(size_t)n * 32 + t] = 0.f;
        fb [(size_t)n * 32 + t] = 0.f;
    }
    if (t < 8) {
        t0[(size_t)n * 8 + t] = 0.f;
        t1[(size_t)n * 8 + t] = 0.f;
        t2[(size_t)n * 8 + t] = 0.f;
    }
}

// Invert the (nbr_in, nbr_out) pair lists into a per-output gather map:
// gidx[k][o] = input index feeding output o through kernel offset k.
// Each output appears at most once per k => unique writes, no atomics.
__global__ void scatter_gidx_kernel(const int* __restrict__ nin,
                                    const int* __restrict__ nout,
                                    int* __restrict__ gidx,
                                    int total, int P, int n) {
    int e = blockIdx.x * blockDim.x + threadIdx.x;
    if (e >= total) return;
    int k = e / P;
    int o = nout[e];
    if (o < n) gidx[(size_t)k * (n + 1) + o] = nin[e];
}

// Repack W_up (8, 64, 32) into a dense (64, 256) B matrix so the whole
// up-projection is one GEMM:  up(8192x256) = x(8192x64) @ W2(64x256).
__global__ void repack_wup_kernel(const float* __restrict__ wup,
                                  float* __restrict__ w2) {
    int e = blockIdx.x * blockDim.x + threadIdx.x;
    if (e >= 64 * 256) return;
    int c = e >> 8;
    int nn = e & 255;
    int kk = nn >> 5;
    int d  = nn & 31;
    w2[e] = wup[(kk * 64 + c) * 32 + d];
}

// ---------------------------------------------------------------------------
// Generic gather-GEMM / sparse-conv kernel on the WMMA pipe.
//   out[m, col0+n] = (res? res[m,col0+n]:0) + bias[n % bias_mod]
//                    + sum_k  f[gidx[k][m]] (1xCIN)  @  W[k] (CINxcout)
// One wave computes a 16(M) x 16(N) f32 tile with V_WMMA_F32_16X16X4_F32.
// Block = 4 waves sharing one N-tile; the W[:, :, n0:n0+16] slice is staged
// in LDS (<=55KB) so B-fragments come from ds_load.
// ---------------------------------------------------------------------------
template <int CIN, int NOFF>
__global__ __launch_bounds__(128) void sconv_wmma_kernel(
    const float* __restrict__ f, int ldf,
    const float* __restrict__ W, int ldw, int cout,
    const float* __restrict__ bias, int bias_mod,
    const int* __restrict__ gidx,
    const float* __restrict__ res,
    float* __restrict__ out, int ldo, int col0,
    int n_rows, int do_relu) {

    __shared__ float sW[NOFF * CIN * 16];

    const int lane = threadIdx.x;             // 0..31
    const int wave = threadIdx.y;             // 0..3
    const int mt   = blockIdx.x * 4 + wave;   // 16-row M tile
    const int n0   = blockIdx.y * 16;         // N tile base (shared by block)

    // Cooperative LDS fill of the weight slice (zero pad beyond cout).
    const int tflat = wave * 32 + lane;
    for (int e = tflat; e < NOFF * CIN * 16; e += 128) {
        int nl = e & 15;
        int rc = e >> 4;                      // k*CIN + c
        int nn = n0 + nl;
        sW[e] = (nn < cout) ? W[(size_t)rc * ldw + nn] : 0.0f;
    }
    __syncthreads();

    const int hi     = lane >> 4;             // 0: K+{0,1}, 1: K+{2,3}
    const int nl     = lane & 15;
    const int m_lane = mt * 16 + nl;          // A-matrix row for this lane
    const int n      = n0 + nl;               // output column for stores

    // C init: column-only bias, identical across the 8 M-rows of a lane.
    float bval = (n < cout) ? bias[((unsigned)n) % (unsigned)bias_mod] : 0.0f;
    v8f acc;
#pragma unroll
    for (int r = 0; r < 8; ++r) acc[r] = bval;

    for (int k = 0; k < NOFF; ++k) {
        int j;
        if (NOFF > 1) {
            j = (m_lane < n_rows) ? gidx[(size_t)k * (n_rows + 1) + m_lane]
                                  : n_rows;    // padding (zero) row
        } else {
            j = (m_lane < n_rows) ? m_lane : 0;
        }
        const float* frow = f + (size_t)j * ldf;
#pragma unroll
        for (int kc = 0; kc < CIN; kc += 4) {
            // A: 16x4 f32; lane holds (M=nl, K=kc+2*hi .. kc+2*hi+1)
            v2f a = *(const v2f*)(frow + kc + 2 * hi);
            // B: 4x16 f32 from LDS, mirrored layout
            v2f b;
            b.x = sW[(k * CIN + kc + 2 * hi    ) * 16 + nl];
            b.y = sW[(k * CIN + kc + 2 * hi + 1) * 16 + nl];
            acc = __builtin_amdgcn_wmma_f32_16x16x4_f32(
                false, a, false, b, (short)0, acc, false, false);
        }
    }

    // D layout: vgpr r, lane -> (M = r + 8*hi, N = nl)
    if (n < cout) {
#pragma unroll
        for (int r = 0; r < 8; ++r) {
            int m = mt * 16 + r + 8 * hi;
            if (m < n_rows) {
                float v = acc[r];
                if (res) v += res[(size_t)m * ldo + col0 + n];
                if (do_relu) v = fmaxf(v, 0.0f);
                out[(size_t)m * ldo + col0 + n] = v;
            }
        }
    }
}

// ---------------------------------------------------------------------------
// Single-block max/argmax (first-max tie-break, like jnp.argmax).
// ---------------------------------------------------------------------------
__global__ __launch_bounds__(256) void argmax_kernel(
    const float* __restrict__ cls, int n, float* __restrict__ red) {
    __shared__ float sv[256];
    __shared__ int   si[256];
    int t = threadIdx.x;
    float best = -INFINITY;
    int   bi   = 0x7fffffff;
    for (int i = t; i < n; i += 256) {
        float v = cls[i];
        if (v > best) { best = v; bi = i; }   // i increasing -> first max kept
    }
    sv[t] = best; si[t] = bi;
    __syncthreads();
    for (int s = 128; s > 0; s >>= 1) {
        if (t < s) {
            if (sv[t + s] > sv[t] ||
                (sv[t + s] == sv[t] && si[t + s] < si[t])) {
                sv[t] = sv[t + s]; si[t] = si[t + s];
            }
        }
        __syncthreads();
    }
    if (t == 0) { red[0] = sv[0]; ((int*)red)[1] = si[0]; }
}

// ---------------------------------------------------------------------------
// Keep mask + pruning + pack the four outputs:
//   [out_pruned (n*32)] [out_cls (n)] [target (n)] [keep (n)]
// ---------------------------------------------------------------------------
__global__ __launch_bounds__(256) void finalize_kernel(
    const float* __restrict__ fcur, const float* __restrict__ cls,
    const int* __restrict__ target, const float* __restrict__ red,
    float* __restrict__ out, int n) {
    int i = blockIdx.x * blockDim.x + threadIdx.x;
    if (i >= n) return;
    float maxv = red[0];
    int   amax = ((const int*)red)[1];
    float s = cls[i];
    bool keep = (s > 0.0f) || (maxv < 0.0f && i == amax);
    float kf = keep ? 1.0f : 0.0f;
#pragma unroll
    for (int c = 0; c < 32; ++c)
        out[(size_t)i * 32 + c] = fcur[(size_t)i * 32 + c] * kf;
    out[(size_t)n * 32 + i] = s;
    out[(size_t)n * 33 + i] = (float)target[i];
    out[(size_t)n * 34 + i] = kf;
}

// ---------------------------------------------------------------------------
// Host launcher
// ---------------------------------------------------------------------------
static inline dim3 grid_for(int rows, int cout) {
    int mt = (rows + 15) / 16;
    return dim3((unsigned)((mt + 3) / 4), (unsigned)((cout + 15) / 16), 1);
}

extern "C" void kernel_launch(void* const* d_in, const int* in_sizes, int n_in,
                              void* d_out, int out_size, void* d_ws, size_t ws_size,
                              hipStream_t stream) {
    // Input order (setup_inputs): x, W_up, b_up, W_conv, b_conv,
    // Wb00, bb00, Wb01, bb01, Wb10, bb10, Wb11, bb11, Wb12, bb12,
    // W_cls, b_cls, nbr_in, nbr_out, target
    const float* x      = (const float*)d_in[0];
    const float* W_up   = (const float*)d_in[1];
    const float* b_up   = (const float*)d_in[2];
    const float* W_conv = (const float*)d_in[3];
    const float* b_conv = (const float*)d_in[4];
    const float* Wb00   = (const float*)d_in[5];
    const float* bb00   = (const float*)d_in[6];
    const float* Wb01   = (const float*)d_in[7];
    const float* bb01   = (const float*)d_in[8];
    const float* Wb10   = (const float*)d_in[9];
    const float* bb10   = (const float*)d_in[10];
    const float* Wb11   = (const float*)d_in[11];
    const float* bb11   = (const float*)d_in[12];
    const float* Wb12   = (const float*)d_in[13];
    const float* bb12   = (const float*)d_in[14];
    const float* W_cls  = (const float*)d_in[15];
    const float* b_cls  = (const float*)d_in[16];
    const int*   nbr_in  = (const int*)d_in[17];
    const int*   nbr_out = (const int*)d_in[18];
    const int*   target  = (const int*)d_in[19];

    const int n    = in_sizes[19];          // n_out (65536)
    const int P    = in_sizes[17] / 27;
    const int N_IN = in_sizes[0] / 64;      // 8192
    const int L    = 3;

    // ---- workspace carving (256B aligned) ----
    char* ws = (char*)d_ws;
    size_t off = 0;
    auto carve = [&](size_t bytes) -> char* {
        off = (off + 255) & ~(size_t)255;
        char* p = ws + off;
        off += bytes;
        return p;
    };
    int*   gidx = (int*)  carve(sizeof(int)   * 27 * (size_t)(n + 1));
    float* fup  = (float*)carve(sizeof(float) * (size_t)(n + 1) * 32);
    float* fa   = (float*)carve(sizeof(float) * (size_t)(n + 1) * 32);
    float* fb   = (float*)carve(sizeof(float) * (size_t)(n + 1) * 32);
    float* t0   = (float*)carve(sizeof(float) * (size_t)(n + 1) * 8);
    float* t1   = (float*)carve(sizeof(float) * (size_t)(n + 1) * 8);
    float* t2   = (float*)carve(sizeof(float) * (size_t)(n + 1) * 8);
    float* cls  = (float*)carve(sizeof(float) * (size_t)n);
    float* wup2 = (float*)carve(sizeof(float) * 64 * 256);
    float* red  = (float*)carve(256);
    (void)ws_size;

    const dim3 blk(32, 4, 1);

    // ---- build inverse neighbor map / zero padding rows ----
    {
        int total = 27 * (n + 1);
        init_gidx_kernel<<<(total + 255) / 256, 256, 0, stream>>>(gidx, total, n);
        zero_pads_kernel<<<1, 64, 0, stream>>>(fup, fa, fb, t0, t1, t2, n);
        int tp = 27 * P;
        scatter_gidx_kernel<<<(tp + 255) / 256, 256, 0, stream>>>(
            nbr_in, nbr_out, gidx, tp, P, n);
        repack_wup_kernel<<<(64 * 256 + 255) / 256, 256, 0, stream>>>(W_up, wup2);
    }

    // ---- up projection: fup = relu(x @ W2 + b_up), viewed (n, 32) ----
    sconv_wmma_kernel<64, 1><<<grid_for(N_IN, 256), blk, 0, stream>>>(
        x, 64, wup2, 256, 256, b_up, 32, nullptr, nullptr,
        fup, 256, 0, N_IN, 1);

    // ---- trunk conv: fa = relu(sconv(fup, W_conv) + b_conv) ----
    sconv_wmma_kernel<32, 27><<<grid_for(n, 32), blk, 0, stream>>>(
        fup, 32, W_conv, 32, 32, b_conv, 32, gidx, nullptr,
        fa, 32, 0, n, 1);

    // ---- residual blocks ----
    float* cur = fa;
    float* nxt = fb;
    for (int l = 0; l < L; ++l) {
        // t0 = relu(sconv(cur, Wb00[l]) + bb00[l])   (32 -> 8)
        sconv_wmma_kernel<32, 27><<<grid_for(n, 8), blk, 0, stream>>>(
            cur, 32, Wb00 + (size_t)l * 27 * 32 * 8, 8, 8, bb00 + l * 8, 8,
            gidx, nullptr, t0, 8, 0, n, 1);
        // nxt[:,0:16] = sconv(t0, Wb01[l]) + bb01[l] + cur[:,0:16]  (8 -> 16)
        sconv_wmma_kernel<8, 27><<<grid_for(n, 16), blk, 0, stream>>>(
            t0, 8, Wb01 + (size_t)l * 27 * 8 * 16, 16, 16, bb01 + l * 16, 16,
            gidx, cur, nxt, 32, 0, n, 0);
        // t1 = relu(cur @ Wb10[l] + bb10[l])         (dense 32 -> 8)
        sconv_wmma_kernel<32, 1><<<grid_for(n, 8), blk, 0, stream>>>(
            cur, 32, Wb10 + (size_t)l * 32 * 8, 8, 8, bb10 + l * 8, 8,
            nullptr, nullptr, t1, 8, 0, n, 1);
        // t2 = relu(sconv(t1, Wb11[l]) + bb11[l])    (8 -> 8)
        sconv_wmma_kernel<8, 27><<<grid_for(n, 8), blk, 0, stream>>>(
            t1, 8, Wb11 + (size_t)l * 27 * 8 * 8, 8, 8, bb11 + l * 8, 8,
            gidx, nullptr, t2, 8, 0, n, 1);
        // nxt[:,16:32] = t2 @ Wb12[l] + bb12[l] + cur[:,16:32] (dense 8 -> 16)
        sconv_wmma_kernel<8, 1><<<grid_for(n, 16), blk, 0, stream>>>(
            t2, 8, Wb12 + (size_t)l * 8 * 16, 16, 16, bb12 + l * 16, 16,
            nullptr, cur, nxt, 32, 16, n, 0);
        float* tmp = cur; cur = nxt; nxt = tmp;
    }

    // ---- classifier: cls = sconv(cur, W_cls) + b_cls   (32 -> 1) ----
    sconv_wmma_kernel<32, 27><<<grid_for(n, 1), blk, 0, stream>>>(
        cur, 32, W_cls, 1, 1, b_cls, 1, gidx, nullptr,
        cls, 1, 0, n, 0);

    // ---- keep mask + pack outputs ----
    argmax_kernel<<<1, 256, 0, stream>>>(cls, n, red);
    finalize_kernel<<<(n + 255) / 256, 256, 0, stream>>>(
        cur, cls, target, red, (float*)d_out, n);
    (void)out_size; (void)n_in;
}